// DiffeomorphicTransform_42391327211583
// MI455X (gfx1250) — compile-verified
//
#include <hip/hip_runtime.h>

// DiffeomorphicTransform (scaling-and-squaring), B=32, H=W=768, C=2, 7 steps.
// Memory-bound gather kernel. MI455X reasoning:
//  - One field = 151 MB < 192 MB global L2: each pass's source (written by the
//    previous pass) is L2-resident, so the 4 data-dependent bilinear gathers
//    per pixel are L2 hits. HBM floor ~ writeback traffic -> ~0.1 ms @ 23.3 TB/s.
//  - Intermediate ping-pong buffers are scratch: store with th:TH_STORE_WB
//    ("force dirty in far cache", overrides default wr-rinse) so their lines
//    stay dirty in L2 and get re-dirtied in place instead of rinsing ~0.9 GB
//    of dead intermediate data to HBM.
//  - Gather addresses are data-dependent (2-deep serial load chain per pixel):
//    each thread processes TWO pixels, and the code is explicitly phased
//    (both center loads -> both address calcs -> all 8 gathers -> lerps ->
//    stores) so the scheduler keeps 2 center loads / 8 gathers in flight
//    instead of serializing the two chains.
//  - wave32: one wave covers a 32-wide row strip -> lane-adjacent gathers land
//    in the same cachelines.
//  - WMMA is inapplicable (no GEMM structure; ~0.5 FLOP/byte) — the optimal
//    instruction mix for this op is vmem-dominated, wmma=0 by design.

#define DH 768
#define DW 768
#define DB 32

typedef __attribute__((ext_vector_type(2))) float v2f;

struct Taps {
    int   i00, i01, i10, i11;   // flat float2 indices of the 4 taps
    float wx, wy;               // interpolation weights
};

// Compute tap indices + weights from a (scaled) flow value at (x, y).
__device__ __forceinline__ Taps make_taps(int base, int x, int y,
                                          float fh, float fw)
{
    // identity grid: linspace(-1,1,N)[i] = -1 + 2*i/(N-1); H == W == 768
    const float step = 2.0f / (float)(DW - 1);
    const float gy = (-1.0f + (float)y * step) + fh;   // channel 0 -> H
    const float gx = (-1.0f + (float)x * step) + fw;   // channel 1 -> W

    // unnormalize, align_corners=False
    const float xs = (gx + 1.0f) * (0.5f * (float)DW) - 0.5f;
    const float ys = (gy + 1.0f) * (0.5f * (float)DH) - 0.5f;
    const float x0f = floorf(xs);
    const float y0f = floorf(ys);

    Taps t;
    t.wx = xs - x0f;
    t.wy = ys - y0f;

    int ix0 = (int)x0f;
    int iy0 = (int)y0f;
    int ix1 = ix0 + 1;
    int iy1 = iy0 + 1;
    // border clamp
    ix0 = min(max(ix0, 0), DW - 1);
    ix1 = min(max(ix1, 0), DW - 1);
    iy0 = min(max(iy0, 0), DH - 1);
    iy1 = min(max(iy1, 0), DH - 1);

    const int r0 = base + iy0 * DW;
    const int r1 = base + iy1 * DW;
    t.i00 = r0 + ix0;
    t.i01 = r0 + ix1;
    t.i10 = r1 + ix0;
    t.i11 = r1 + ix1;
    return t;
}

__device__ __forceinline__ v2f bilerp(float2 g00, float2 g01,
                                      float2 g10, float2 g11,
                                      float wx, float wy)
{
    // reference form: top = a*(1-wx)+b*wx ; out = top*(1-wy)+bot*wy
    const float owx = 1.0f - wx;
    const float owy = 1.0f - wy;
    const float tx = g00.x * owx + g01.x * wx;
    const float ty = g00.y * owx + g01.y * wx;
    const float bx = g10.x * owx + g11.x * wx;
    const float by = g10.y * owx + g11.y * wx;
    v2f s;
    s.x = tx * owy + bx * wy;
    s.y = ty * owy + by * wy;
    return s;
}

__device__ __forceinline__ void store_l2_wb(float2* dst, int idx, v2f o)
{
    // th:TH_STORE_WB = keep dirty in far cache (L2), override default wr-rinse.
    asm volatile("global_store_b64 %0, %1, off th:TH_STORE_WB"
                 :: "v"((unsigned long long)(size_t)(dst + idx)), "v"(o));
}

__global__ __launch_bounds__(256) void diffeo_warp_step(
    const float2* __restrict__ src, float2* __restrict__ dst, float scale)
{
    const int x  = blockIdx.x * 32 + threadIdx.x;
    const int y0 = blockIdx.y * 16 + threadIdx.y;   // this thread: rows y0, y0+8
    const int y1 = y0 + 8;
    const int b  = blockIdx.z;
    const int base = b * (DH * DW);

    const int idx0 = base + y0 * DW + x;
    const int idx1 = base + y1 * DW + x;

    // ---- phase 1: both center loads in flight ----
    float2 f0 = src[idx0];
    float2 f1 = src[idx1];

    // fold the 1/2^T pre-scale into pass 1
    f0.x *= scale; f0.y *= scale;
    f1.x *= scale; f1.y *= scale;

    // ---- phase 2: both address calcs ----
    const Taps t0 = make_taps(base, x, y0, f0.x, f0.y);
    const Taps t1 = make_taps(base, x, y1, f1.x, f1.y);

    // gfx1250: global_prefetch_b8 the second gather rows (no LOADcnt cost,
    // speculative-safe) while the first rows' loads fly.
    __builtin_prefetch((const void*)(src + t0.i10), 0, 1);
    __builtin_prefetch((const void*)(src + t1.i10), 0, 1);

    // ---- phase 3: all 8 gathers in flight ----
    const float2 a00 = src[t0.i00];
    const float2 a01 = src[t0.i01];
    const float2 a10 = src[t0.i10];
    const float2 a11 = src[t0.i11];
    const float2 b00 = src[t1.i00];
    const float2 b01 = src[t1.i01];
    const float2 b10 = src[t1.i10];
    const float2 b11 = src[t1.i11];

    // ---- phase 4: lerps; gathers were unscaled -> scale once (bilerp linear)
    const v2f s0 = bilerp(a00, a01, a10, a11, t0.wx, t0.wy);
    const v2f s1 = bilerp(b00, b01, b10, b11, t1.wx, t1.wy);

    v2f o0, o1;
    o0.x = f0.x + s0.x * scale;
    o0.y = f0.y + s0.y * scale;
    o1.x = f1.x + s1.x * scale;
    o1.y = f1.y + s1.y * scale;

    // ---- phase 5: stores (L2-pinned) ----
    store_l2_wb(dst, idx0, o0);
    store_l2_wb(dst, idx1, o1);
}

extern "C" void kernel_launch(void* const* d_in, const int* in_sizes, int n_in,
                              void* d_out, int out_size, void* d_ws, size_t ws_size,
                              hipStream_t stream) {
    (void)in_sizes; (void)n_in; (void)out_size; (void)ws_size;

    const float2* in  = (const float2*)d_in[0];
    float2*       out = (float2*)d_out;
    float2*       ws  = (float2*)d_ws;   // needs >= 32*768*768*8 = 151 MB

    const dim3 block(32, 8, 1);
    const dim3 grid(DW / 32, DH / 16, DB);   // 2 rows per thread

    const float s = 1.0f / 128.0f;  // 1 / 2^TIME_STEP

    // 7 passes; odd count -> schedule so the last write lands in d_out:
    // in -> out -> ws -> out -> ws -> out -> ws -> out
    diffeo_warp_step<<<grid, block, 0, stream>>>(in,  out, s);     // pass 1 (scaled)
    diffeo_warp_step<<<grid, block, 0, stream>>>(out, ws,  1.0f);  // pass 2
    diffeo_warp_step<<<grid, block, 0, stream>>>(ws,  out, 1.0f);  // pass 3
    diffeo_warp_step<<<grid, block, 0, stream>>>(out, ws,  1.0f);  // pass 4
    diffeo_warp_step<<<grid, block, 0, stream>>>(ws,  out, 1.0f);  // pass 5
    diffeo_warp_step<<<grid, block, 0, stream>>>(out, ws,  1.0f);  // pass 6
    diffeo_warp_step<<<grid, block, 0, stream>>>(ws,  out, 1.0f);  // pass 7
}